// FNO_Blocks_24687472018089
// MI455X (gfx1250) — compile-verified
//
#include <hip/hip_runtime.h>
#include <math.h>

// FNO blocks for MI455X (gfx1250, wave32).
// Spectral conv decomposed into mode-restricted DFT GEMMs; the two large
// GEMMs (forward DFT over w; fused c2r-IDFT + bypass linear + GELU) use
// V_WMMA_F32_16X16X4_F32. Middle stages (DFT_h / mode-mix / IDFT_h) are
// tiny (~1.2 GFLOP/layer) and run as VALU kernels.
//
// Round-1 fix: twiddle B-fragments are now a single branchless ds_load from
// a combined 512-entry table (cos | -sin), removing the exec-mask divergence
// the compiler generated for the two-array ternary in the WMMA inner loops.
// The T[k=0]=1 special case is folded away by pre-halving Zr[kx=0] in K4.
//
// Workspace layout (floats), total 9,437,184 f = 36 MiB:
//   U   [B][CW][H][32]   (n<16: Re, n>=16: Im of kx=0..15)   4,194,304
//   Vr,Vi  [B][CW][32m][16kx]                                  262,144 each
//   V2r,V2i[B][CW][32m][16kx]                                  262,144 each
//   Z   [B][H][CW][32]   (k<16: Re, k>=16: Im; Zr[0] halved) 4,194,304

#define TWO_PI 6.283185307179586f

typedef float v2f __attribute__((ext_vector_type(2)));
typedef float v8f __attribute__((ext_vector_type(8)));

static __device__ __forceinline__ v8f wmma4_f32(v2f a, v2f b, v8f c) {
  return __builtin_amdgcn_wmma_f32_16x16x4_f32(false, a, false, b,
                                               (short)0, c, false, false);
}

// ---------------------------------------------------------------------------
// K1: U[row][n] = sum_w X[row][w] * Bm[w][n], row=(b,c,h) in [0,131072)
//     Bm[w][n] = n<16 ? cos(2pi n w/256) : -sin(2pi (n-16) w/256)  (rfft sign)
// grid 2048 x 256 threads (8 waves); block tile: 64 rows x 32 cols.
// ---------------------------------------------------------------------------
__global__ void k1_dftw(const float* __restrict__ X, float* __restrict__ U) {
  __shared__ float Bt[512];               // [0..255]=cos, [256..511]=-sin
  int tid = threadIdx.x;
  { float sv, cv; __sincosf(TWO_PI * (float)tid * (1.0f/256.0f), &sv, &cv);
    Bt[tid] = cv; Bt[tid + 256] = -sv; }
  __syncthreads();

  int wave = tid >> 5, lane = tid & 31;
  int rt = wave >> 1, nt = wave & 1;
  long r0 = (long)blockIdx.x * 64 + rt * 16;
  int l15 = lane & 15;
  int kk  = (lane >> 4) << 1;
  int n   = nt * 16 + l15;
  int nn  = n & 15;
  int ofs = (n & 16) << 4;                // 0 for Re cols, 256 for Im cols

  const float* xr = X + (r0 + l15) * 256;
  int a0 = ((nn * kk) & 255) + ofs;       // angle index for b.x
  int a1 = ((nn * (kk + 1)) & 255) + ofs; // angle index for b.y
  int stp = (nn * 4) & 255;               // angle increment per k0 step

  v8f acc = {0,0,0,0,0,0,0,0};
  #pragma unroll 4
  for (int k0 = 0; k0 < 256; k0 += 4) {
    v2f a = *(const v2f*)(xr + k0 + kk);
    v2f b = { Bt[a0], Bt[a1] };
    acc = wmma4_f32(a, b, acc);
    a0 = ((a0 + stp) & 255) + ofs - (a0 & 256 ? 0 : 0); // keep ofs in index
    a0 = ((a0 - ofs) & 255) + ofs;                      // (folded below)
    a1 = ((a1 - ofs + stp) & 255) + ofs;
  }
  long mbase = r0 + ((lane >> 4) << 3);   // M = r + 8*(lane>=16)
  #pragma unroll
  for (int r = 0; r < 8; ++r)
    U[(mbase + r) * 32 + n] = acc[r];
}

// ---------------------------------------------------------------------------
// K2: DFT over h, 32 ky modes (0..15 and 240..255), e^{-i phi}.
//     Vr + iVi = sum_h (Ur + iUi)(cos - i sin)
// grid 512=(b,c) x 256 threads.
// ---------------------------------------------------------------------------
__global__ void k2_dfth(const float* __restrict__ U,
                        float* __restrict__ Vr, float* __restrict__ Vi) {
  __shared__ float Us[256 * 32];
  __shared__ float ct[256], st[256];
  int tid = threadIdx.x;
  long bc = blockIdx.x;
  { float sv, cv; __sincosf(TWO_PI * (float)tid * (1.0f/256.0f), &sv, &cv);
    ct[tid] = cv; st[tid] = sv; }
  const float* up = U + bc * 8192;
  for (int j = 0; j < 32; ++j) Us[tid + j * 256] = up[tid + j * 256];
  __syncthreads();

  int kx = tid & 15, mq = tid >> 4;       // mq in [0,16): handles m=mq, m=mq+16
  int ky2 = 240 + mq;
  float vr1 = 0, vi1 = 0, vr2 = 0, vi2 = 0;
  for (int h = 0; h < 256; ++h) {
    float ur = Us[h * 32 + kx], ui = Us[h * 32 + 16 + kx];
    int t1 = (mq  * h) & 255;
    int t2 = (ky2 * h) & 255;
    float c1 = ct[t1], s1 = st[t1], c2 = ct[t2], s2 = st[t2];
    vr1 += ur * c1 + ui * s1;  vi1 += ui * c1 - ur * s1;
    vr2 += ur * c2 + ui * s2;  vi2 += ui * c2 - ur * s2;
  }
  long base = bc * 512;
  Vr[base + mq * 16 + kx]        = vr1;  Vi[base + mq * 16 + kx]        = vi1;
  Vr[base + (mq + 16) * 16 + kx] = vr2;  Vi[base + (mq + 16) * 16 + kx] = vi2;
}

// ---------------------------------------------------------------------------
// K3: per-mode channel mix. V'[b,o] = sum_i V[b,i] * w[i,o]  (complex)
//     m<16 -> w1[.,.,m,kx], m>=16 -> w2[.,.,m-16,kx]
// grid 512=(m,kx) x 256 threads; each thread: one o, two b.
// ---------------------------------------------------------------------------
__global__ void k3_mix(const float* __restrict__ Vr, const float* __restrict__ Vi,
                       const float* __restrict__ w1r, const float* __restrict__ w1i,
                       const float* __restrict__ w2r, const float* __restrict__ w2i,
                       float* __restrict__ V2r, float* __restrict__ V2i) {
  __shared__ float Vsr[512], Vsi[512];
  int tid  = threadIdx.x;
  int mode = blockIdx.x;
  int m = mode >> 4, kx = mode & 15;
  for (int p = tid; p < 512; p += 256) {
    int b = p >> 6, i = p & 63;
    long idx = ((long)(b * 64 + i) * 32 + m) * 16 + kx;
    Vsr[p] = Vr[idx];  Vsi[p] = Vi[idx];
  }
  __syncthreads();

  int o = tid & 63, bq = tid >> 6;        // bq in [0,4): handles b=bq, bq+4
  const float* wr = (m < 16) ? w1r : w2r;
  const float* wi = (m < 16) ? w1i : w2i;
  int mm = m & 15;
  long woff = (long)o * 256 + mm * 16 + kx;
  float ar0 = 0, ai0 = 0, ar1 = 0, ai1 = 0;
  for (int i = 0; i < 64; ++i) {
    float WR = wr[(long)i * 16384 + woff];
    float WI = wi[(long)i * 16384 + woff];
    float vr0 = Vsr[bq * 64 + i],       vi0 = Vsi[bq * 64 + i];
    float vr1 = Vsr[(bq + 4) * 64 + i], vi1 = Vsi[(bq + 4) * 64 + i];
    ar0 += vr0 * WR - vi0 * WI;  ai0 += vr0 * WI + vi0 * WR;
    ar1 += vr1 * WR - vi1 * WI;  ai1 += vr1 * WI + vi1 * WR;
  }
  long ob0 = ((long)(bq * 64 + o) * 32 + m) * 16 + kx;
  long ob1 = ((long)((bq + 4) * 64 + o) * 32 + m) * 16 + kx;
  V2r[ob0] = ar0;  V2i[ob0] = ai0;
  V2r[ob1] = ar1;  V2i[ob1] = ai1;
}

// ---------------------------------------------------------------------------
// K4: inverse DFT over h (e^{+i phi}), fold 1/(H*W).  Z[b][h][o][k32]
//     Zr[kx=0] is additionally halved so K5 can use a uniform 2*cos column.
// grid 512=(b,o) x 256 threads (t=h).
// ---------------------------------------------------------------------------
__global__ void k4_idfth(const float* __restrict__ V2r, const float* __restrict__ V2i,
                         float* __restrict__ Z) {
  __shared__ float Wr[512], Wi[512];
  __shared__ float ct[256], st[256];
  int tid = threadIdx.x;
  int b = blockIdx.x >> 6, o = blockIdx.x & 63;
  { float sv, cv; __sincosf(TWO_PI * (float)tid * (1.0f/256.0f), &sv, &cv);
    ct[tid] = cv; st[tid] = sv; }
  long base = (long)(b * 64 + o) * 512;
  for (int p = tid; p < 512; p += 256) { Wr[p] = V2r[base + p]; Wi[p] = V2i[base + p]; }
  __syncthreads();

  int h = tid;
  float zr[16], zi[16];
  #pragma unroll
  for (int k = 0; k < 16; ++k) { zr[k] = 0.0f; zi[k] = 0.0f; }
  for (int m = 0; m < 32; ++m) {
    int ky = (m < 16) ? m : 224 + m;      // 240..255
    int t = (ky * h) & 255;
    float c = ct[t], s = st[t];
    #pragma unroll
    for (int kx = 0; kx < 16; ++kx) {
      float vr = Wr[m * 16 + kx], vi = Wi[m * 16 + kx];
      zr[kx] += vr * c - vi * s;
      zi[kx] += vr * s + vi * c;
    }
  }
  const float sc = 1.0f / 65536.0f;       // 1/(H*W)
  zr[0] *= 0.5f;                          // fold T[k=0,w]=1 into Z (table has 2cos)
  float* zp = Z + ((long)(b * 256 + h) * 64 + o) * 32;
  #pragma unroll
  for (int kx = 0; kx < 16; ++kx) { zp[kx] = zr[kx] * sc; zp[16 + kx] = zi[kx] * sc; }
}

// ---------------------------------------------------------------------------
// K5: fused c2r IDFT over w + bypass linear + bias + exact GELU.
//   Y[o,w] = gelu( sum_k Z[o,k]*T[k,w] + sum_i bw[o,i]*X[i,w] + bb[o] )
//   T[k,w] = Tt[((k&15)*w)&255 + ((k&16)<<4)],  Tt = [2cos | -2sin]
//   (T[0,w]: Zr[0] pre-halved in K4; T[16,w] = -2sin(0) = 0 naturally.)
// grid 4096=(b,h,half) x 256 threads (8 waves); block tile 64o x 128w.
// Stages its X slice in LDS before storing -> safe for in-place X==Y.
// ---------------------------------------------------------------------------
__global__ void k5_out(const float* __restrict__ X, const float* __restrict__ Z,
                       const float* __restrict__ bw, const float* __restrict__ bb,
                       float* __restrict__ Y) {
  __shared__ float Xs[64 * 128];
  __shared__ float Zs[64 * 32];
  __shared__ float Bws[64 * 64];
  __shared__ float biasS[64];
  __shared__ float Tt[512];               // [0..255]=2cos, [256..511]=-2sin
  int tid = threadIdx.x;
  int bi = blockIdx.x;
  int b = bi >> 9, h = (bi >> 1) & 255, half = bi & 1;
  int wbase = half * 128;
  { float sv, cv; __sincosf(TWO_PI * (float)tid * (1.0f/256.0f), &sv, &cv);
    Tt[tid] = 2.0f * cv; Tt[tid + 256] = -2.0f * sv; }
  // stage X[b, :, h, wbase..wbase+127]  (coalesced)
  for (int p = tid; p < 8192; p += 256) {
    int i = p >> 7, wv = p & 127;
    Xs[p] = X[((long)(b * 64 + i) * 256 + h) * 256 + wbase + wv];
  }
  { const float* zp = Z + (long)(b * 256 + h) * 2048;
    for (int p = tid; p < 2048; p += 256) Zs[p] = zp[p]; }
  for (int p = tid; p < 4096; p += 256) Bws[p] = bw[p];
  if (tid < 64) biasS[tid] = bb[tid];
  __syncthreads();

  int wave = tid >> 5, lane = tid & 31;
  int kk  = (lane >> 4) << 1;
  int l15 = lane & 15;
  int mofs = (lane >> 4) << 3;
  for (int q = 0; q < 4; ++q) {
    int tileid = wave * 4 + q;            // 32 tiles: 4 o-tiles x 8 w-tiles
    int ot = tileid >> 3, wt = tileid & 7;
    int oa   = ot * 16 + l15;
    int wloc = wt * 16 + l15;
    int w    = wbase + wloc;
    v8f acc = {0,0,0,0,0,0,0,0};
    // spectral c2r: K = 32  (Zr|Zi)
    #pragma unroll
    for (int k0 = 0; k0 < 32; k0 += 4) {
      int ka = k0 + kk;
      int kb = ka + 1;
      int ia = (((ka & 15) * w) & 255) + ((ka & 16) << 4);
      int ib = (((kb & 15) * w) & 255) + ((kb & 16) << 4);
      v2f a = { Zs[oa * 32 + ka], Zs[oa * 32 + kb] };
      v2f bf = { Tt[ia], Tt[ib] };
      acc = wmma4_f32(a, bf, acc);
    }
    // bypass linear: K = 64 (input channels)
    #pragma unroll
    for (int k0 = 0; k0 < 64; k0 += 4) {
      v2f a  = { Bws[oa * 64 + k0 + kk], Bws[oa * 64 + k0 + kk + 1] };
      v2f bf = { Xs[(k0 + kk) * 128 + wloc], Xs[(k0 + kk + 1) * 128 + wloc] };
      acc = wmma4_f32(a, bf, acc);
    }
    // epilogue: bias + exact-erf GELU, store
    #pragma unroll
    for (int r = 0; r < 8; ++r) {
      int oe = ot * 16 + r + mofs;
      float v = acc[r] + biasS[oe];
      float g = 0.5f * v * (1.0f + erff(v * 0.70710678118654752f));
      Y[((long)(b * 64 + oe) * 256 + h) * 256 + wbase + wt * 16 + l15] = g;
    }
  }
}

// ---------------------------------------------------------------------------
extern "C" void kernel_launch(void* const* d_in, const int* in_sizes, int n_in,
                              void* d_out, int out_size, void* d_ws, size_t ws_size,
                              hipStream_t stream) {
  (void)in_sizes; (void)n_in; (void)out_size; (void)ws_size;
  const float* x   = (const float*)d_in[0];
  const float* w1r = (const float*)d_in[1];
  const float* w1i = (const float*)d_in[2];
  const float* w2r = (const float*)d_in[3];
  const float* w2i = (const float*)d_in[4];
  const float* bw  = (const float*)d_in[5];
  const float* bb  = (const float*)d_in[6];

  float* ws  = (float*)d_ws;
  float* U   = ws;                 // 4,194,304
  float* Vr  = U   + 4194304;      //   262,144
  float* Vi  = Vr  + 262144;
  float* V2r = Vi  + 262144;
  float* V2i = V2r + 262144;
  float* Zb  = V2i + 262144;       // 4,194,304   (total 36 MiB)

  float* Yout = (float*)d_out;
  const long WMODE = 64L * 64 * 16 * 16;   // per-layer complex-weight plane

  for (int l = 0; l < 4; ++l) {
    const float* Xin = (l == 0) ? x : (const float*)d_out;
    const float* w1r_l = w1r + (long)l * WMODE;
    const float* w1i_l = w1i + (long)l * WMODE;
    const float* w2r_l = w2r + (long)l * WMODE;
    const float* w2i_l = w2i + (long)l * WMODE;
    const float* bw_l  = bw  + (long)l * 4096;
    const float* bb_l  = bb  + (long)l * 64;

    k1_dftw <<<2048, 256, 0, stream>>>(Xin, U);
    k2_dfth <<< 512, 256, 0, stream>>>(U, Vr, Vi);
    k3_mix  <<< 512, 256, 0, stream>>>(Vr, Vi, w1r_l, w1i_l, w2r_l, w2i_l, V2r, V2i);
    k4_idfth<<< 512, 256, 0, stream>>>(V2r, V2i, Zb);
    // In-place safe for l>=1: each block stages its X slice in LDS before
    // writing, and read/write footprints per block are identical & disjoint.
    k5_out  <<<4096, 256, 0, stream>>>(Xin, Zb, bw_l, bb_l, Yout);
  }
}